// PopulationWarmup_11519102288081
// MI455X (gfx1250) — compile-verified
//
#include <hip/hip_runtime.h>
#include <math.h>

// ---------------------------------------------------------------------------
// PopulationWarmup on gfx1250 (MI455X):
//   - single persistent kernel over the 365 sequential steps (sync-latency
//     bound: ~730 grid barriers dominate; compute ~1.5 GFLOP, traffic ~650 MB)
//   - scan carry (X_p, X_h, X_y) held in registers (fixed thread<->element map)
//   - one float segment-sum per step via global_atomic_add_f32 (D_X==0 makes
//     the cnt/M_inf segment-sum dead code: log(0/x + 1) == 0)
//   - CDNA5 async global->LDS double-buffering of the eps[t] row
//     (global_load_async_to_lds_b64 + s_wait_asynccnt)
//   - all large scratch lives in the (dead-until-the-end) d_out regions;
//     d_ws usage is 8 bytes (barrier counters)
// ---------------------------------------------------------------------------

#define NIND   32768         // individuals (== grid threads)
#define NINF   65536         // infections
#define NSTEP  365
#define DT     5
#define NBLK   128
#define TPB    256
#define INF_PER_BLK (NINF / NBLK)   // 512 infections per block
// thread l owns infections j = blk*512 + 2l, 2l+1  (contiguous -> b64 async)

typedef __attribute__((address_space(3))) char lds_char_t;

__device__ __forceinline__ void async_copy_b64_to_lds(const float* gsrc, float* ldst) {
  unsigned loff = (unsigned)(unsigned long long)(lds_char_t*)ldst;  // LDS byte offset
  unsigned long long gaddr = (unsigned long long)gsrc;
  asm volatile("global_load_async_to_lds_b64 %0, %1, off"
               :: "v"(loff), "v"(gaddr) : "memory");
}

__device__ __forceinline__ void wait_async0() {
  asm volatile("s_wait_asynccnt 0x0" ::: "memory");
}

__device__ __forceinline__ void grid_barrier(unsigned* cnt, unsigned* gen) {
  __threadfence();           // release: flush this thread's global writes to L2
  __syncthreads();
  if (threadIdx.x == 0) {
    unsigned g = __hip_atomic_load(gen, __ATOMIC_RELAXED, __HIP_MEMORY_SCOPE_AGENT);
    unsigned a = __hip_atomic_fetch_add(cnt, 1u, __ATOMIC_ACQ_REL, __HIP_MEMORY_SCOPE_AGENT);
    if (a == (unsigned)(NBLK - 1)) {
      __hip_atomic_store(cnt, 0u, __ATOMIC_RELAXED, __HIP_MEMORY_SCOPE_AGENT);
      __hip_atomic_store(gen, g + 1u, __ATOMIC_RELEASE, __HIP_MEMORY_SCOPE_AGENT);
    } else {
      while (__hip_atomic_load(gen, __ATOMIC_RELAXED, __HIP_MEMORY_SCOPE_AGENT) == g)
        __builtin_amdgcn_s_sleep(1);
    }
  }
  __syncthreads();
  __threadfence();           // acquire: invalidate L0 so new L2 data is seen
}

// ---------------------------------------------------------------------------
__global__ void pw_init(float* __restrict__ Ybuf, unsigned* cnt, unsigned* gen) {
  int i = blockIdx.x * blockDim.x + threadIdx.x;
  int stride = gridDim.x * blockDim.x;
  for (int s = i; s < 2 * NIND; s += stride) Ybuf[s] = 0.0f;
  if (i == 0) { *cnt = 0u; *gen = 0u; }
}

// ---------------------------------------------------------------------------
__global__ __launch_bounds__(TPB)
void pw_sim(const float* __restrict__ EIR,    const float* __restrict__ log_d,
            const float* __restrict__ tau0_g, const float* __restrict__ taumax_g,
            const float* __restrict__ eps,    const int* __restrict__ idx_g,
            const int*   __restrict__ offsets,
            unsigned* cnt, unsigned* gen,
            float* __restrict__ out /* [Y(N) | X_y(M) | X_h(N)] */) {
  const int l  = threadIdx.x;
  const int b  = blockIdx.x;
  const int i  = b * TPB + l;                 // this thread's individual
  const int jb = b * INF_PER_BLK + 2 * l;     // this thread's 2 infections

  // scratch aliased onto the output buffer (fully rewritten at the end):
  float* coefA  = out;                        // [N]  D_h * D_m   (Y region)
  float* Ybuf   = out + NIND;                 // [2][N]           (X_y region)
  float* sigAct = out + NIND + NINF;          // [N]  +-sigma_y   (X_h region)

  __shared__ float ebuf[2][INF_PER_BLK];      // double-buffered eps row chunk

  // ---- per-individual register state ----
  const int off = offsets[i];
  float Xp = 0.0f, Xh = 0.0f;

  // ---- per-infection register state / constants ----
  float t0[2], tmx[2], aa[2], bb[2], ldi[2], Xy[2], yk[2];
  int   id[2];
  bool  mm[2];
#pragma unroll
  for (int k = 0; k < 2; ++k) {
    int j = jb + k;
    float tt0 = tau0_g[j], tM = taumax_g[j];
    float dur = tM - tt0;
    float a   = fminf(0.018f * dur, 4.4f);
    float c   = a / (1.0f + dur * (1.0f / 35.0f));
    t0[k] = tt0; tmx[k] = tM; aa[k] = a;
    bb[k] = logf(a / c) / dur;
    int ii = idx_g[j];
    id[k] = ii; ldi[k] = log_d[ii];
    Xy[k] = 0.0f; yk[k] = 0.0f; mm[k] = false;
  }

  // prefetch eps row 0 (each lane copies its own 8B -> its own LDS slot)
  async_copy_b64_to_lds(eps + jb, &ebuf[0][2 * l]);

  for (int t = 0; t < NSTEP; ++t) {
    const int   cur = t & 1, nxt = cur ^ 1;
    const int   t_int = t * DT;
    const float tf = (float)t_int;

    // ---------------- phase A: per-individual ----------------
    {
      int  d   = t_int - off;
      bool act = d >= 0;
      int  rel = act ? (d / DT) : 0;          // age index, in [0, 364]
      // eff_EIR[rel] computed inline: EIR[rel % 73] * body_surface(5*rel)/A_MAX
      float trel = (float)(rel * DT);
      float age  = fminf(trel, 7300.0f) * (1.0f / 365.0f);
      float bs   = 2.009715297378839e-1f +
                   age * (1.7984405384216584e-1f +
                   age * (-2.121587990857579e-2f +
                   age * (1.3554013908681017e-3f +
                   age * (-2.847697353524861e-5f))));
      float E  = act ? (EIR[rel % 73] * bs * (1.0f / 1.63f)) : 0.0f;
      float S1 = 0.049f + 0.951f / (1.0f + E * (1.0f / 0.032f));
      float S2 = 0.14f  + 0.86f  / (1.0f + powf(Xp * (1.0f / 1514.4f), 2.04f));
      float h  = S1 * S2 * E;
      // immunity factors use X_h BEFORE the update (as in reference)
      float sig = 0.8124038404635961f * rsqrtf(1.0f + Xh * (1.0f / 4.8f)); // sqrt(0.66)
      float Dh  = 1.0f / (1.0f + Xh * (1.0f / 97.3f));
      float Dm  = 1.0f - 0.9f * expf(-trel * (0.69314718055994531f / (365.0f * 2.53f)));
      coefA[i]  = Dh * Dm;
      sigAct[i] = act ? sig : -sig;           // sign bit encodes act
      Xp += E;
      Xh += h;
    }

    grid_barrier(cnt, gen);                    // A -> B (coef visible)

    // eps row t is needed now; row t+1 goes in flight behind the compute
    wait_async0();
    if (t + 1 < NSTEP)
      async_copy_b64_to_lds(eps + (size_t)(t + 1) * NINF + jb, &ebuf[nxt][2 * l]);

    // ---------------- phase B: per-infection ----------------
    float* Yc = Ybuf + (size_t)cur * NIND;
#pragma unroll
    for (int k = 0; k < 2; ++k) {
      float e_t = ebuf[cur][2 * l + k];
      float cf  = coefA[id[k]];
      float sg  = sigAct[id[k]];
      bool  m   = (t0[k] <= tf) && (tmx[k] > tf) && (sg > 0.0f);
      mm[k] = m;
      if (m) {
        float Dy   = 1.0f / (1.0f + Xy[k] * (1.0f / 3.5f));
        float lnyG = aa[k] * expf(-bb[k] * (tf - t0[k])) - 4.8f;
        if (lnyG <= 0.0f) lnyG = 1e-10f;
        float Elny = Dy * cf * (ldi[k] + lnyG);       // log_term == 0 (D_X==0)
        float y    = expf(Elny + sg * e_t);
        yk[k] = y;
        atomicAdd(&Yc[id[k]], y);
      } else {
        yk[k] = 0.0f;
      }
    }

    grid_barrier(cnt, gen);                    // B -> C (Y complete)

    // ---------------- phase C: cross-immunity + next-buffer zero ----------
#pragma unroll
    for (int k = 0; k < 2; ++k)
      if (mm[k]) Xy[k] += Yc[id[k]] - yk[k];
    Ybuf[(size_t)nxt * NIND + i] = 0.0f;       // safe: nxt buffer idle here
  }

  // final Y is Ybuf[364 & 1 == 0]; read it before output stores clobber it
  float Yfin = Ybuf[i];
  grid_barrier(cnt, gen);                      // all scratch reads complete

  out[i]                 = Yfin;               // Y      (overwrites coefA)
  out[NIND + NINF + i]   = Xh;                 // X_h    (overwrites sigAct)
  out[NIND + jb]         = Xy[0];              // X_y    (overwrites Ybuf)
  out[NIND + jb + 1]     = Xy[1];
}

// ---------------------------------------------------------------------------
extern "C" void kernel_launch(void* const* d_in, const int* in_sizes, int n_in,
                              void* d_out, int out_size, void* d_ws, size_t ws_size,
                              hipStream_t stream) {
  (void)in_sizes; (void)n_in; (void)out_size; (void)ws_size;

  const float* EIR    = (const float*)d_in[0];
  const float* log_d  = (const float*)d_in[1];
  const float* tau0   = (const float*)d_in[2];
  const float* tauM   = (const float*)d_in[3];
  const float* eps    = (const float*)d_in[4];
  const int*   idx    = (const int*)d_in[5];
  const int*   offs   = (const int*)d_in[6];

  float*    out  = (float*)d_out;              // [Y(N) | X_y(M) | X_h(N)]
  float*    Ybuf = out + NIND;                 // scratch alias (X_y region)
  unsigned* cnt  = (unsigned*)d_ws;            // 8 bytes of workspace total
  unsigned* gen  = cnt + 1;

  hipLaunchKernelGGL(pw_init, dim3(64), dim3(256), 0, stream, Ybuf, cnt, gen);
  hipLaunchKernelGGL(pw_sim, dim3(NBLK), dim3(TPB), 0, stream,
                     EIR, log_d, tau0, tauM, eps, idx, offs, cnt, gen, out);
}